// ZICrossEntropy_68341519614312
// MI455X (gfx1250) — compile-verified
//
#include <hip/hip_runtime.h>

typedef __attribute__((ext_vector_type(4))) float f4;
typedef __attribute__((ext_vector_type(2))) float v2f;
typedef __attribute__((ext_vector_type(8))) float v8f;

#define B_      32
#define C_      8
#define H_      256
#define W_      256
#define PLANE   (H_*W_)      /* 65536 floats per channel plane */
#define GW      512          /* gt width */
#define NBLOCKS 1024
#define NTHREADS 256

// ---------------------------------------------------------------------------
// Kernel 1: streaming pass. Each thread-iteration handles 4 consecutive x
// pixels of one (b, y) row: 8 x float4 NT loads from the channel planes plus
// 4 x float4 NT loads from the 2x2 GT blocks (48 B/pixel, the roofline
// minimum). Branchless binning + exp2/log2 logsumexp, per-block deterministic
// reduction to d_ws.
// ---------------------------------------------------------------------------
__global__ __launch_bounds__(NTHREADS)
void ce_partial(const float* __restrict__ logits,
                const float* __restrict__ gt,
                float* __restrict__ partial) {
    const int tid      = blockIdx.x * NTHREADS + threadIdx.x;
    const int nthreads = NBLOCKS * NTHREADS;              // 262144
    const int NGROUP   = B_ * H_ * (W_ / 4);              // 524288 -> 2 iters/thread

    const float lo[8] = {1.f, 2.f, 3.f, 4.f, 6.f, 9.f, 13.f, 17.f};
    const float hi[8] = {1.f, 2.f, 3.f, 5.f, 8.f, 12.f, 16.f, 100.f};

    float acc = 0.f;

    for (int q = tid; q < NGROUP; q += nthreads) {
        const int x4 = q & 63;            // W/4 = 64 groups per row
        const int y  = (q >> 6) & 255;
        const int b  = q >> 14;

        // ---- logits: 8 channel planes, float4 each, non-temporal ----
        const f4* Lp = (const f4*)(logits + (size_t)b * C_ * PLANE + (size_t)y * W_ + (size_t)x4 * 4);
        f4 l[8];
#pragma unroll
        for (int c = 0; c < 8; ++c)
            l[c] = __builtin_nontemporal_load(Lp + (size_t)c * (PLANE / 4));

        // ---- GT 2x2 blocks: rows 2y and 2y+1, 8 floats each ----
        const float* g0 = gt + ((size_t)b * GW + (size_t)(2 * y)) * GW + (size_t)x4 * 8;
        f4 r0a = __builtin_nontemporal_load((const f4*)g0);
        f4 r0b = __builtin_nontemporal_load((const f4*)g0 + 1);
        f4 r1a = __builtin_nontemporal_load((const f4*)(g0 + GW));
        f4 r1b = __builtin_nontemporal_load((const f4*)(g0 + GW) + 1);

        const float gsum[4] = { r0a.x + r0a.y + r1a.x + r1a.y,
                                r0a.z + r0a.w + r1a.z + r1a.w,
                                r0b.x + r0b.y + r1b.x + r1b.y,
                                r0b.z + r0b.w + r1b.z + r1b.w };

#pragma unroll
        for (int j = 0; j < 4; ++j) {
            const float g = gsum[j];

            // branchless bin classification (bins are disjoint)
            int cls = 0;
#pragma unroll
            for (int i = 0; i < 8; ++i)
                cls = (g >= lo[i] && g <= hi[i]) ? (i + 1) : cls;

            float xs[8];
#pragma unroll
            for (int c = 0; c < 8; ++c) xs[c] = l[c][j];

            float m = xs[0];
#pragma unroll
            for (int c = 1; c < 8; ++c) m = fmaxf(m, xs[c]);

            float s = 0.f;
#pragma unroll
            for (int c = 0; c < 8; ++c)
                s += __builtin_amdgcn_exp2f((xs[c] - m) * 1.4426950408889634f);

            const float lse = m + 0.69314718055994531f * __builtin_amdgcn_logf(s);

            const int label = (cls > 0) ? (cls - 1) : 0;
            float picked = xs[0];
#pragma unroll
            for (int c = 1; c < 8; ++c) picked = (label == c) ? xs[c] : picked;

            acc += (cls > 0) ? (lse - picked) : 0.f;
        }
    }

    // ---- deterministic block reduction: 256 -> 32 via LDS, 32 -> 1 via shfl
    __shared__ float sred[NTHREADS];
    sred[threadIdx.x] = acc;
    __syncthreads();
    if (threadIdx.x < 32) {
        float v = 0.f;
#pragma unroll
        for (int i = 0; i < NTHREADS / 32; ++i) v += sred[threadIdx.x + i * 32];
#pragma unroll
        for (int off = 16; off > 0; off >>= 1) v += __shfl_xor(v, off, 32);
        if (threadIdx.x == 0) partial[blockIdx.x] = v;
    }
}

// ---------------------------------------------------------------------------
// Kernel 2: single-wave finisher. Folds 1024 block partials through
// V_WMMA_F32_16X16X4_F32 used as a 64-way adder tree: D = A x ones + C.
// With B = all-ones, D[m][n] = sum_k A[m][k] for every n, so chaining C
// accumulates the total; sum_m D[m][0] = sum of all packed A values.
// Packing is an arbitrary bijection of 64 partials onto the 16x4 A tile
// (a.x = p[base+lane], a.y = p[base+32+lane]), which is valid because we
// sum every slot. Uniform control flow -> EXEC all-ones as WMMA requires.
// ---------------------------------------------------------------------------
__global__ __launch_bounds__(32)
void ce_final(const float* __restrict__ partial, float* __restrict__ out, int n) {
    const int lane = threadIdx.x;

    v2f ones; ones.x = 1.f; ones.y = 1.f;
    v8f c = {};

    for (int base = 0; base < n; base += 64) {
        v2f a;
        a.x = partial[base + lane];
        a.y = partial[base + 32 + lane];
        // 8 args: (neg_a, A, neg_b, B, c_mod, C, reuse_a, reuse_b)
        c = __builtin_amdgcn_wmma_f32_16x16x4_f32(
                /*neg_a=*/false, a, /*neg_b=*/false, ones,
                /*c_mod=*/(short)0, c, /*reuse_a=*/false, /*reuse_b=*/false);
    }

    // lanes 0-15 hold rows M=0..7 of column N=lane, lanes 16-31 rows M=8..15.
    float v = c[0] + c[1] + c[2] + c[3] + c[4] + c[5] + c[6] + c[7];
    v += __shfl_xor(v, 16, 32);          // add M=0..7 and M=8..15 halves
    if (lane == 0) out[0] = v * (1.0f / (float)B_);
}

extern "C" void kernel_launch(void* const* d_in, const int* in_sizes, int n_in,
                              void* d_out, int out_size, void* d_ws, size_t ws_size,
                              hipStream_t stream) {
    const float* logits = (const float*)d_in[0];   // (32, 8, 256, 256) fp32
    const float* gt     = (const float*)d_in[1];   // (32, 1, 512, 512) fp32
    float* out     = (float*)d_out;                // scalar loss
    float* partial = (float*)d_ws;                 // 1024 floats of scratch

    ce_partial<<<NBLOCKS, NTHREADS, 0, stream>>>(logits, gt, partial);
    ce_final<<<1, 32, 0, stream>>>(partial, out, NBLOCKS);
}